// TrafficAugmentation_71322226917643
// MI455X (gfx1250) — compile-verified
//
#include <hip/hip_runtime.h>

namespace {
constexpr int   FPB   = 32;        // flows per block == threads per block (one wave32)
constexpr int   TILE  = 64;        // S-positions staged per tile
constexpr int   RS    = TILE + 4;  // padded LDS row stride (floats): 272B, 16B-aligned, bank-skewed
constexpr int   SPOS  = 2048;      // S (positions per flow), fixed by the reference
constexpr float kMSS  = 1448.0f;
constexpr int   BUFN  = FPB * RS;            // floats per LDS buffer
constexpr unsigned BUFB = (unsigned)BUFN * 4; // bytes per LDS buffer
}

// ---- CDNA5 async global->LDS copies (per-lane, tracked by ASYNCcnt) ----
// GVS form: mem = SGPR_U64 + VGPR_I32 (+0), LDS dest = per-lane VGPR byte address.
__device__ __forceinline__ void async_b128_s(unsigned lds_byte, unsigned voff_byte,
                                             unsigned long long sbase) {
  asm volatile("global_load_async_to_lds_b128 %0, %1, %2"
               :: "v"(lds_byte), "v"(voff_byte), "s"(sbase)
               : "memory");
}
// GV form (64-bit per-lane address) for the rare partial-block path.
__device__ __forceinline__ void async_b128_v(unsigned lds_byte, const float* gptr) {
  asm volatile("global_load_async_to_lds_b128 %0, %1, off"
               :: "v"(lds_byte), "v"((unsigned long long)(const void*)gptr)
               : "memory");
}
__device__ __forceinline__ void wait_async0() {
  asm volatile("s_wait_asynccnt 0" ::: "memory");
}
__device__ __forceinline__ void wait_ds0() {
  asm volatile("s_wait_dscnt 0" ::: "memory");
}

// rem must be a separate mul+sub (no FMA) to match JAX at the rem>0 boundary.
__device__ __forceinline__ float split_rem(float buf, float nf_f) {
#pragma clang fp contract(off)
  float prod = nf_f * kMSS;
  return buf - prod;
}

__global__ __launch_bounds__(FPB) void traffic_sim_kernel(
    const float* __restrict__ x, const float* __restrict__ dl,
    const float* __restrict__ rt, float* __restrict__ out, int B) {
  constexpr int S = SPOS;
  __shared__ __align__(16) float sx[2 * BUFN];
  __shared__ __align__(16) float sd[2 * BUFN];
  __shared__ __align__(16) float sr[2 * BUFN];

  const int    tid     = threadIdx.x;
  const int    flow0   = blockIdx.x * FPB;
  const int    flow    = flow0 + tid;
  const bool   valid   = (flow < B);
  const size_t rowbase = (size_t)flow * (size_t)S;

  // Lane map for staging: p4 constant per lane, fl = 2*i + hi (affine in i).
  const int hi  = tid >> 4;
  const int p4f = (tid & 15) << 2;                       // float index within row
  const unsigned ldsLane = (unsigned)((hi * RS + p4f) * 4); // LDS byte offset of lane
  const unsigned voffLane = (unsigned)((hi * S + p4f) * 4); // per-lane global byte offset

  // low 32 bits of a generic pointer to __shared__ == LDS byte offset
  const unsigned bx0 = (unsigned)(unsigned long long)(&sx[0]);
  const unsigned bd0 = (unsigned)(unsigned long long)(&sd[0]);
  const unsigned br0 = (unsigned)(unsigned long long)(&sr[0]);

  const unsigned long long sbx = (unsigned long long)(const void*)(x  + (size_t)flow0 * S);
  const unsigned long long sbd = (unsigned long long)(const void*)(dl + (size_t)flow0 * S);
  const unsigned long long sbr = (unsigned long long)(const void*)(rt + (size_t)flow0 * S);

  const bool fullBlock = (flow0 + FPB <= B);

  auto stage = [&](int j0, int cb) {
    unsigned lx = bx0 + (unsigned)cb * BUFB + ldsLane;
    unsigned ld_ = bd0 + (unsigned)cb * BUFB + ldsLane;
    unsigned lr = br0 + (unsigned)cb * BUFB + ldsLane;
    if (fullBlock) {
      unsigned vo = voffLane + (unsigned)(j0 * 4);
#pragma unroll
      for (int i = 0; i < 16; ++i) {
        async_b128_s(lx, vo, sbx);
        async_b128_s(ld_, vo, sbd);
        async_b128_s(lr, vo, sbr);
        vo += (unsigned)(2 * S * 4);      // next pair of rows (constant: 16384)
        lx += (unsigned)(2 * RS * 4);     // 544
        ld_ += (unsigned)(2 * RS * 4);
        lr += (unsigned)(2 * RS * 4);
      }
    } else {                              // partial last block: clamp flow index
#pragma unroll
      for (int i = 0; i < 16; ++i) {
        int fl = 2 * i + hi;
        int fs = flow0 + fl; if (fs >= B) fs = B - 1;
        size_t g = (size_t)fs * (size_t)S + (size_t)(j0 + p4f);
        unsigned lo = (unsigned)((fl * RS + p4f) * 4) + (unsigned)cb * BUFB;
        async_b128_v(bx0 + lo, x + g);
        async_b128_v(bd0 + lo, dl + g);
        async_b128_v(br0 + lo, rt + g);
      }
    }
  };

  bool  active = valid, need_rtt = true;
  float rtt = 0.0f, buf = 0.0f;
  int   w = 0;  // output write cursor
  int   cb = 0;

  stage(0, 0);  // preload tile 0

  for (int j0 = 0; j0 < S; j0 += TILE) {
    wait_async0();                     // tile j0 resident in buffer cb
    const int  jn       = j0 + TILE;
    const bool lastTile = (jn >= S);
    if (!lastTile) stage(jn, cb ^ 1);  // overlap next tile's DMA with compute

    const float* px = &sx[cb * BUFN + tid * RS];
    const float* pd = &sd[cb * BUFN + tid * RS];
    const float* pr = &sr[cb * BUFN + tid * RS];

    for (int t4 = 0; t4 < TILE; t4 += 4) {
      float4 x4 = *(const float4*)(px + t4);   // ds_load_b128, 16B-aligned
      float4 d4 = *(const float4*)(pd + t4);
      float4 r4 = *(const float4*)(pr + t4);
      float xa[4] = {x4.x, x4.y, x4.z, x4.w};
      float da[4] = {d4.x, d4.y, d4.z, d4.w};
      float ra[4] = {r4.x, r4.y, r4.z, r4.w};
#pragma unroll
      for (int u = 0; u < 4; ++u) {
        const float xj = xa[u], dj = da[u], rj = ra[u];
        const bool  last = lastTile && (t4 + u == TILE - 1);
        // --- branchless automaton step (mirrors the JAX scan) ---
        const bool nr_old    = need_rtt;
        const bool start_new = active && nr_old;
        const bool stop      = start_new && (xj <= 0.0f);
        active = active && !stop;
        const float rb = (start_new && !stop) ? rj : rtt;  // fresh RTT draw
        rtt = active ? (rb - dj) : rb;                     // consume packet
        buf = active ? (buf + xj) : buf;
        const bool flush = active && ((rtt <= 0.0f) || last);
        need_rtt = flush || (!active && nr_old);
        if (flush) {                                       // only divergent part
          float nf_f = fmaxf(ceilf(buf / kMSS) - 1.0f, 0.0f);
          float rem  = split_rem(buf, nf_f);
          int   nf   = (int)nf_f;
          int   lim  = nf < (S - w) ? nf : (S - w);
          for (int k = 0; k < lim; ++k) out[rowbase + w + k] = kMSS + 40.0f;
          w += lim;
          if (rem > 0.0f && w < S) { out[rowbase + w] = rem + 40.0f; ++w; }
          buf = 0.0f;
        }
      }
    }
    wait_ds0();   // drain DS reads of buffer cb before it is async-overwritten
    cb ^= 1;
  }

  // zero-fill the tail (harness poisons d_out; reference pads with 0)
  if (valid) {
    for (int p = w; p < S; ++p) out[rowbase + p] = 0.0f;
  }
}

extern "C" void kernel_launch(void* const* d_in, const int* in_sizes, int n_in,
                              void* d_out, int out_size, void* d_ws, size_t ws_size,
                              hipStream_t stream) {
  (void)n_in; (void)d_ws; (void)ws_size; (void)out_size;
  const float* x      = (const float*)d_in[0];
  const float* delays = (const float*)d_in[1];
  const float* rtts   = (const float*)d_in[2];
  float*       out    = (float*)d_out;

  const int B = in_sizes[0] / SPOS;   // 2048

  dim3 block(FPB);
  dim3 grid((B + FPB - 1) / FPB);
  traffic_sim_kernel<<<grid, block, 0, stream>>>(x, delays, rtts, out, B);
}